// Loss_comb2_44040594653652
// MI455X (gfx1250) — compile-verified
//
#include <hip/hip_runtime.h>
#include <hip/hip_bf16.h>

typedef __attribute__((ext_vector_type(2))) float v2f;
typedef __attribute__((ext_vector_type(8))) float v8f;

#define GRID_A 512
#define BLK_A  256
#define BLK_B  256
#define NSLOTS (GRID_A + 1)

// ---------------------------------------------------------------- helpers

__device__ __forceinline__ float sigmoidf_(float x) {
    return 1.0f / (1.0f + __expf(-x));
}

// numerically-stable softplus(x) = log(1 + e^x)
__device__ __forceinline__ float softplusf_(float x) {
    return fmaxf(x, 0.0f) + log1pf(__expf(-fabsf(x)));
}

// Exact f32 sum across all 32 lanes of the wave using V_WMMA_F32_16X16X4_F32.
// A (16x4): lane L<16 holds A[L][0]=s, A[L][1]=0; lane L>=16 holds A[L-16][2]=s,
// A[L-16][3]=0.  B = all ones => D[m][n] = s(m) + s(m+16) for every column.
// Per-lane fold of the 8 accumulator VGPRs sums M=0..7 (lanes 0-15) or
// M=8..15 (lanes 16-31); one shfl_xor(16) add completes the 32-lane sum.
__device__ __forceinline__ float wave_sum32_wmma(float s) {
    v2f a; a[0] = s;    a[1] = 0.0f;
    v2f b; b[0] = 1.0f; b[1] = 1.0f;
    v8f c = {0.f, 0.f, 0.f, 0.f, 0.f, 0.f, 0.f, 0.f};
    c = __builtin_amdgcn_wmma_f32_16x16x4_f32(
            /*neg_a=*/false, a, /*neg_b=*/false, b,
            /*c_mod=*/(short)0, c, /*reuse_a=*/false, /*reuse_b=*/false);
    float t = ((c[0] + c[1]) + (c[2] + c[3])) + ((c[4] + c[5]) + (c[6] + c[7]));
    t += __shfl_xor(t, 16, 32);
    return t;
}

// ---------------------------------------------------------------- kernel A
// Dense negative loss over both levels.  float4 streaming, grid-stride.
// ws slot layout per block b: [pos, neg, reg, cnt_pos, cnt_neg, reg_w]
__global__ void neg_loss_kernel(const float* __restrict__ cls0,
                                const float* __restrict__ prob_fine,
                                const float* __restrict__ cls1,
                                const float* __restrict__ prob_coarse,
                                long n0q, long n1q,
                                float* __restrict__ ws) {
    float accN = 0.0f, accC = 0.0f;
    const long total  = n0q + n1q;
    const long stride = (long)gridDim.x * blockDim.x;

    for (long i = (long)blockIdx.x * blockDim.x + threadIdx.x; i < total; i += stride) {
        const float4* lp;
        const float4* pp;
        float nf;
        long  j;
        if (i < n0q) { lp = (const float4*)cls0; pp = (const float4*)prob_fine;   nf = 2.0f; j = i; }
        else         { lp = (const float4*)cls1; pp = (const float4*)prob_coarse; nf = 1.0f; j = i - n0q; }

        // speculative prefetch of next grid-stride iteration (global_prefetch_b8)
        __builtin_prefetch((const void*)(lp + j + stride), 0, 1);
        __builtin_prefetch((const void*)(pp + j + stride), 0, 1);

        float4 l = lp[j];
        float4 p = pp[j];

        float lx[4] = {l.x, l.y, l.z, l.w};
        float px[4] = {p.x, p.y, p.z, p.w};
#pragma unroll
        for (int e = 0; e < 4; ++e) {
            float x    = lx[e];
            float mask = (px[e] == -1.0f) ? 1.0f : 0.0f;
            float w    = sigmoidf_(x) * mask;       // stop_grad(sigmoid) * negmask
            accC += w;                              // cnt_neg
            accN += softplusf_(x) * w * nf;         // nll * w * nf
        }
    }

    __shared__ float sN[BLK_A];
    __shared__ float sC[BLK_A];
    const int tid = threadIdx.x;
    sN[tid] = accN;
    sC[tid] = accC;
    __syncthreads();
    for (int s = BLK_A / 2; s > 0; s >>= 1) {
        if (tid < s) { sN[tid] += sN[tid + s]; sC[tid] += sC[tid + s]; }
        __syncthreads();
    }
    if (tid == 0) {
        float* slot = ws + (long)blockIdx.x * 6;
        slot[0] = 0.0f;   // pos
        slot[1] = sN[0];  // neg
        slot[2] = 0.0f;   // reg
        slot[3] = 0.0f;   // cnt_pos
        slot[4] = sC[0];  // cnt_neg
        slot[5] = 0.0f;   // reg_w
    }
}

// ---------------------------------------------------------------- kernel B
// Sparse positive loss + L1 regression loss (768 items each). Single block.
__global__ void sparse_loss_kernel(const float* __restrict__ cls0,
                                   const float* __restrict__ cls1,
                                   const float* __restrict__ reg0,
                                   const float* __restrict__ reg1,
                                   const int*  __restrict__ cpf,  // coord_prob_fine   (4,128,4)
                                   const int*  __restrict__ cpc,  // coord_prob_coarse (4, 64,4)
                                   const int*  __restrict__ cdf,  // coord_diff_fine   (4,128,4)
                                   const int*  __restrict__ cdc,  // coord_diff_coarse (4, 64,4)
                                   const float* __restrict__ df,  // diff_fine   (4,128,6)
                                   const float* __restrict__ dc,  // diff_coarse (4, 64,6)
                                   float* __restrict__ ws) {
    const int NPF = 4 * 128;          // fine items
    const int NPC = 4 * 64;           // coarse items
    const int NT  = NPF + NPC;

    float accP = 0.0f, accCP = 0.0f, accR = 0.0f, accRW = 0.0f;

    // ---- positive loss (gather from cls logits) ----
    for (int i = threadIdx.x; i < NT; i += blockDim.x) {
        const float* logit;
        const int*   cp;
        int K, D, b, k;
        float pf;
        if (i < NPF) { logit = cls0; cp = cpf; K = 128; D = 96; pf = 2.0f; b = i / 128;        k = i % 128; }
        else         { int i2 = i - NPF;
                       logit = cls1; cp = cpc; K = 64;  D = 48; pf = 1.0f; b = i2 / 64;        k = i2 % 64; }
        const int* c = cp + ((long)b * K + k) * 4;
        float valid = (c[0] > -1) ? 1.0f : 0.0f;
        int c0 = max(c[0], 0), c1 = max(c[1], 0), c2 = max(c[2], 0), c3 = max(c[3], 0);
        long idx = ((((long)b * 2 + c0) * D + c1) * D + c2) * D + c3;
        float lp = logit[idx];
        float wp = (1.0f - sigmoidf_(lp)) * valid;   // stop_grad(1 - sigmoid) * valid
        accP  += softplusf_(-lp) * wp * pf;          // -log_sigmoid(lp)*wp*w2*pf, w2==1
        accCP += wp;
    }

    // ---- L1 regression loss (gather from reg tensors) ----
    for (int i = threadIdx.x; i < NT; i += blockDim.x) {
        const float* regp;
        const int*   cd;
        const float* dgt;
        int K, D, b, k;
        if (i < NPF) { regp = reg0; cd = cdf; dgt = df; K = 128; D = 96; b = i / 128; k = i % 128; }
        else         { int i2 = i - NPF;
                       regp = reg1; cd = cdc; dgt = dc; K = 64;  D = 48; b = i2 / 64; k = i2 % 64; }
        const int* c = cd + ((long)b * K + k) * 4;
        float valid = (c[0] > -1) ? 1.0f : 0.0f;
        int c0 = max(c[0], 0), c1 = max(c[1], 0), c2 = max(c[2], 0), c3 = max(c[3], 0);
        long dd   = (long)D * D * D;
        long base = (long)b * 12 * dd;
        long off  = (((long)c1 * D + c2) * D + c3);
#pragma unroll
        for (int j = 0; j < 6; ++j) {
            float pred = regp[base + (long)(j * 2 + c0) * dd + off];
            float g    = dgt[((long)b * K + k) * 6 + j];
            accR += fabsf(pred - g) * valid;
        }
        accRW += valid;
    }

    __shared__ float sP[BLK_B], sCP[BLK_B], sR[BLK_B], sRW[BLK_B];
    const int tid = threadIdx.x;
    sP[tid] = accP; sCP[tid] = accCP; sR[tid] = accR; sRW[tid] = accRW;
    __syncthreads();
    for (int s = BLK_B / 2; s > 0; s >>= 1) {
        if (tid < s) {
            sP[tid]  += sP[tid + s];
            sCP[tid] += sCP[tid + s];
            sR[tid]  += sR[tid + s];
            sRW[tid] += sRW[tid + s];
        }
        __syncthreads();
    }
    if (tid == 0) {
        float* slot = ws + (long)GRID_A * 6;
        slot[0] = sP[0];   // pos
        slot[1] = 0.0f;    // neg
        slot[2] = sR[0];   // reg
        slot[3] = sCP[0];  // cnt_pos
        slot[4] = 0.0f;    // cnt_neg
        slot[5] = sRW[0];  // reg_w
    }
}

// ---------------------------------------------------------------- kernel C
// One wave32: serial per-lane accumulation over slots, then six exact-f32
// cross-lane sums via v_wmma_f32_16x16x4_f32.
__global__ void final_reduce_kernel(const float* __restrict__ ws, int nslots,
                                    float* __restrict__ out) {
    const int lane = threadIdx.x;  // blockDim.x == 32, one wave
    float acc[6] = {0.f, 0.f, 0.f, 0.f, 0.f, 0.f};
    for (int i = lane; i < nslots; i += 32) {
        const float* slot = ws + (long)i * 6;
#pragma unroll
        for (int q = 0; q < 6; ++q) acc[q] += slot[q];
    }
    // Full EXEC here (loop has reconverged) — WMMA legal.
    float t0 = wave_sum32_wmma(acc[0]);
    float t1 = wave_sum32_wmma(acc[1]);
    float t2 = wave_sum32_wmma(acc[2]);
    float t3 = wave_sum32_wmma(acc[3]);
    float t4 = wave_sum32_wmma(acc[4]);
    float t5 = wave_sum32_wmma(acc[5]);
    if (lane == 0) {
        out[0] = t0;  // loss:   pos
        out[1] = t1;  //         neg
        out[2] = t2;  //         reg
        out[3] = t3;  // weight: cnt_pos
        out[4] = t4;  //         cnt_neg
        out[5] = t5;  //         reg_w
    }
}

// ---------------------------------------------------------------- launch

extern "C" void kernel_launch(void* const* d_in, const int* in_sizes, int n_in,
                              void* d_out, int out_size, void* d_ws, size_t ws_size,
                              hipStream_t stream) {
    (void)in_sizes; (void)n_in; (void)out_size; (void)ws_size;

    const float* cls0        = (const float*)d_in[0];   // (4,2,96,96,96)
    const float* reg0        = (const float*)d_in[1];   // (4,12,96,96,96)
    const float* cls1        = (const float*)d_in[2];   // (4,2,48,48,48)
    const float* reg1        = (const float*)d_in[3];   // (4,12,48,48,48)
    const float* prob_coarse = (const float*)d_in[4];   // (4,2,48,48,48)
    const float* prob_fine   = (const float*)d_in[5];   // (4,2,96,96,96)
    const int*   cpc         = (const int*)d_in[6];     // coord_prob_coarse
    const int*   cpf         = (const int*)d_in[7];     // coord_prob_fine
    const int*   cdc         = (const int*)d_in[8];     // coord_diff_coarse
    const int*   cdf         = (const int*)d_in[9];     // coord_diff_fine
    const float* dc          = (const float*)d_in[10];  // diff_coarse
    const float* df          = (const float*)d_in[11];  // diff_fine

    float* out = (float*)d_out;
    float* ws  = (float*)d_ws;

    const long n0q = (long)4 * 2 * 96 * 96 * 96 / 4;  // float4 chunks, level 0
    const long n1q = (long)4 * 2 * 48 * 48 * 48 / 4;  // float4 chunks, level 1

    neg_loss_kernel<<<GRID_A, BLK_A, 0, stream>>>(cls0, prob_fine, cls1, prob_coarse,
                                                  n0q, n1q, ws);
    sparse_loss_kernel<<<1, BLK_B, 0, stream>>>(cls0, cls1, reg0, reg1,
                                                cpf, cpc, cdf, cdc, df, dc, ws);
    final_reduce_kernel<<<1, 32, 0, stream>>>(ws, NSLOTS, out);
}